// CosineLoss_7284264534292
// MI455X (gfx1250) — compile-verified
//
#include <hip/hip_runtime.h>

#define N_ROWS 4096
#define N_COLS 1024

typedef __attribute__((ext_vector_type(2))) float v2f;
typedef __attribute__((ext_vector_type(8))) float v8f;

// d_ws float layout: [0,1024) s-accumulator | [1024] diag count | [1025,1025+4096) rnorm

__global__ __launch_bounds__(256)
void CosLoss_rownorm(const float* __restrict__ feats,
                     float* __restrict__ rnorm,
                     float* __restrict__ diag) {
    __shared__ float red[256];
    const int row = blockIdx.x;
    const float4* rp = (const float4*)(feats + (size_t)row * N_COLS);
    float4 v = rp[threadIdx.x];                       // 256 threads * 4 = 1024 cols
    float ss = v.x * v.x + v.y * v.y + v.z * v.z + v.w * v.w;
    red[threadIdx.x] = ss;
    __syncthreads();
    for (int off = 128; off > 0; off >>= 1) {
        if (threadIdx.x < off) red[threadIdx.x] += red[threadIdx.x + off];
        __syncthreads();
    }
    if (threadIdx.x == 0) {
        float s2 = red[0];
        rnorm[row] = (s2 > 0.0f) ? (1.0f / sqrtf(s2)) : 0.0f;
        if (s2 > 0.0f) atomicAdd(diag, 1.0f);         // sum_i ||u_i||^2 (exact diag term)
    }
}

// One wave per (16-column slab, 256-row chunk). A = all-ones 16x4 f32 (layout
// invariant), B = 4 rows x 16 cols of rnorm-scaled features. C accumulates the
// per-column partial sums across the K loop via V_WMMA_F32_16X16X4_F32.
__global__ __launch_bounds__(32)
void CosLoss_colsum_wmma(const float* __restrict__ feats,
                         const float* __restrict__ rnorm,
                         float* __restrict__ s) {
    const int lane  = threadIdx.x;        // 0..31, wave32
    const int col   = lane & 15;          // B/C column = lane mod 16
    const int half  = lane >> 4;          // lane half selects row within VGPR
    const int cbase = blockIdx.x * 16;
    int row         = blockIdx.y * (N_ROWS / 16);
    const int rowEnd = row + (N_ROWS / 16);

    v8f c = {};
    v2f a;
    a[0] = 1.0f;
    a[1] = 1.0f;

    for (; row < rowEnd; row += 4) {
        const int r0 = row + half;            // rows {0,1} of the K=4 chunk
        const int r1 = row + 2 + half;        // rows {2,3}
        v2f b;
        b[0] = feats[(size_t)r0 * N_COLS + cbase + col] * rnorm[r0];
        b[1] = feats[(size_t)r1 * N_COLS + cbase + col] * rnorm[r1];
        // (neg_a, A, neg_b, B, c_mod, C, reuse_a, reuse_b)
        c = __builtin_amdgcn_wmma_f32_16x16x4_f32(false, a, false, b,
                                                  (short)0, c, false, false);
    }
    // C layout: VGPR0, lanes 0-15 = M=0 row, N=lane. All M rows are identical
    // (A rows identical), so lanes 0-15 carry the 16 column partial sums.
    if (lane < 16) atomicAdd(&s[cbase + lane], c[0]);
}

__global__ __launch_bounds__(256)
void CosLoss_finalize(const float* __restrict__ s,
                      const float* __restrict__ diag,
                      float* __restrict__ out) {
    __shared__ float red[256];
    float4 v = ((const float4*)s)[threadIdx.x];       // 256*4 = 1024
    red[threadIdx.x] = v.x * v.x + v.y * v.y + v.z * v.z + v.w * v.w;
    __syncthreads();
    for (int off = 128; off > 0; off >>= 1) {
        if (threadIdx.x < off) red[threadIdx.x] += red[threadIdx.x + off];
        __syncthreads();
    }
    if (threadIdx.x == 0) {
        const double P = (double)N_ROWS * (double)(N_ROWS - 1) * 0.5;
        double sum_offdiag = 0.5 * ((double)red[0] - (double)diag[0]);
        out[0] = (float)(1.0 - sum_offdiag / P);
    }
}

extern "C" void kernel_launch(void* const* d_in, const int* in_sizes, int n_in,
                              void* d_out, int out_size, void* d_ws, size_t ws_size,
                              hipStream_t stream) {
    const float* feats = (const float*)d_in[0];
    float* ws    = (float*)d_ws;
    float* s     = ws;            // 1024 floats
    float* diag  = ws + 1024;     // 1 float
    float* rnorm = ws + 1025;     // 4096 floats

    // Zero the atomic accumulators every call (graph-capture-safe memset node).
    hipMemsetAsync(ws, 0, 1025 * sizeof(float), stream);

    CosLoss_rownorm<<<N_ROWS, 256, 0, stream>>>(feats, rnorm, diag);

    dim3 grid(N_COLS / 16, 16);   // 64 column slabs x 16 row chunks
    CosLoss_colsum_wmma<<<grid, 32, 0, stream>>>(feats, rnorm, s);

    CosLoss_finalize<<<1, 256, 0, stream>>>(s, diag, (float*)d_out);
}